// ChamferLoss_81003083203707
// MI455X (gfx1250) — compile-verified
//
#include <hip/hip_runtime.h>
#include <math.h>

// Shapes fixed by the reference: B=4, D=20, H=W=128.
#define BLOCK 256
constexpr int Dn   = 20;
constexpr int HW   = 128 * 128;     // per-batch spatial size
constexpr int CHW  = 60 * HW;       // per-batch stride for pred (3*D ch) and gt (D*3 ch)
constexpr int BPB  = HW / BLOCK;    // blocks per batch = 64 (power of two)

// ---------- wave32 reductions (gfx1250 is wave32: explicit width=32) ----------
__device__ __forceinline__ float wred_max(float v) {
#pragma unroll
    for (int o = 16; o > 0; o >>= 1) v = fmaxf(v, __shfl_down(v, o, 32));
    return v;
}
__device__ __forceinline__ float wred_sum(float v) {
#pragma unroll
    for (int o = 16; o > 0; o >>= 1) v += __shfl_down(v, o, 32);
    return v;
}

// Block reductions: 8 waves -> LDS -> wave0. Result valid in thread 0.
// Helpers leave `red` reusable (trailing barrier), so order is fixed & deterministic.
__device__ __forceinline__ float block_max(float v, float* red) {
    const int lane = threadIdx.x & 31, wid = threadIdx.x >> 5;
    v = wred_max(v);
    if (lane == 0) red[wid] = v;
    __syncthreads();
    float r = -__builtin_inff();
    if (wid == 0) {
        r = (lane < (BLOCK / 32)) ? red[lane] : -__builtin_inff();
        r = wred_max(r);
    }
    __syncthreads();
    return r;
}
__device__ __forceinline__ float block_sum(float v, float* red) {
    const int lane = threadIdx.x & 31, wid = threadIdx.x >> 5;
    v = wred_sum(v);
    if (lane == 0) red[wid] = v;
    __syncthreads();
    float r = 0.0f;
    if (wid == 0) {
        r = (lane < (BLOCK / 32)) ? red[lane] : 0.0f;
        r = wred_sum(r);
    }
    __syncthreads();
    return r;
}

// Scalar-friendly voxel decomposition: each 256-thread block sits inside one batch.
__device__ __forceinline__ void voxel_coords(int& k, int& b, int& hw) {
    const int blk = blockIdx.x;
    b  = blk / BPB;                              // uniform per block (SALU)
    hw = (blk % BPB) * BLOCK + threadIdx.x;      // contiguous across the wave
    k  = blk * BLOCK + threadIdx.x;
}

// ---------- Pass 1: per-block max of L1 pairwise distance over valid voxels ----------
__global__ void k_dist_max(const float* __restrict__ pred, const float* __restrict__ gt,
                           const float* __restrict__ gmask, float* __restrict__ blockmax) {
    __shared__ float red[BLOCK / 32];
    int k, b, hw;
    voxel_coords(k, b, hw);
    const float* pb = pred + (size_t)b * CHW + hw;
    const float* gb = gt   + (size_t)b * CHW + hw;
    // gfx1250 prefetch path; locality 3 -> WGP scope: fills all cache levels here,
    // and the 192MB L2 keeps everything resident for pass 2.
    __builtin_prefetch(pb, 0, 3);
    __builtin_prefetch(gb, 0, 3);

    float dst[Dn][3];
#pragma unroll
    for (int j = 0; j < Dn; j++) {
        dst[j][0] = gb[(size_t)(j * 3 + 0) * HW];
        dst[j][1] = gb[(size_t)(j * 3 + 1) * HW];
        dst[j][2] = gb[(size_t)(j * 3 + 2) * HW];
    }
    float m = -__builtin_inff();
    for (int i = 0; i < Dn; i++) {
        const float s0 = pb[(size_t)(0 * Dn + i) * HW];
        const float s1 = pb[(size_t)(1 * Dn + i) * HW];
        const float s2 = pb[(size_t)(2 * Dn + i) * HW];
#pragma unroll
        for (int j = 0; j < Dn; j++) {
            const float d = fabsf(s0 - dst[j][0]) + fabsf(s1 - dst[j][1]) + fabsf(s2 - dst[j][2]);
            m = fmaxf(m, d);
        }
    }
    if (!(gmask[k] > 0.0f)) m = -__builtin_inff();
    m = block_max(m, red);
    if (threadIdx.x == 0) blockmax[blockIdx.x] = m;
}

// ---------- Pass 1b: reduce block maxes -> max_d (single block, fixed order) ----------
__global__ void k_max_reduce(const float* __restrict__ blockmax, int nblk,
                             float* __restrict__ maxd) {
    __shared__ float red[BLOCK / 32];
    float v = -__builtin_inff();
    for (int i = threadIdx.x; i < nblk; i += BLOCK) v = fmaxf(v, blockmax[i]);
    v = block_max(v, red);
    if (threadIdx.x == 0) maxd[0] = v;
}

// ---------- Pass 2: fused chamfer mins + smooth-L1, per-block partial sums ----------
__global__ void k_main(const float* __restrict__ pred, const float* __restrict__ gt,
                       const float* __restrict__ cmask, const float* __restrict__ gmask,
                       const float* __restrict__ pnum, const float* __restrict__ gnum,
                       const float* __restrict__ maxd_p, int nblk,
                       float* __restrict__ part) {
    __shared__ float red[BLOCK / 32];
    const float maxd = maxd_p[0];
    int k, b, hw;
    voxel_coords(k, b, hw);
    const float* pb = pred  + (size_t)b * CHW + hw;
    const float* gb = gt    + (size_t)b * CHW + hw;
    const float* cb = cmask + (size_t)b * Dn * HW + hw;

    float dst[Dn][3], cm[Dn], cmm[Dn], d2s[Dn];
#pragma unroll
    for (int j = 0; j < Dn; j++) {
        dst[j][0] = gb[(size_t)(j * 3 + 0) * HW];
        dst[j][1] = gb[(size_t)(j * 3 + 1) * HW];
        dst[j][2] = gb[(size_t)(j * 3 + 2) * HW];
        cm[j]  = cb[(size_t)j * HW];
        cmm[j] = cm[j] * maxd;       // cm in {0,1}: dist - cmm[j] == reference term exactly
        d2s[j] = __builtin_inff();
    }

    float s2d_acc = 0.0f;
    for (int i = 0; i < Dn; i++) {
        const float s0 = pb[(size_t)(0 * Dn + i) * HW];
        const float s1 = pb[(size_t)(1 * Dn + i) * HW];
        const float s2 = pb[(size_t)(2 * Dn + i) * HW];
        float smin = __builtin_inff();
#pragma unroll
        for (int j = 0; j < Dn; j++) {
            const float d = fabsf(s0 - dst[j][0]) + fabsf(s1 - dst[j][1]) + fabsf(s2 - dst[j][2]);
            smin   = fminf(smin, d - cmm[j]);   // s2d inner: min_j(dist - cm_j*max_d)
            d2s[j] = fminf(d2s[j], d);          // d2s: min_i(dist)
        }
        s2d_acc += smin;
    }

    const float vw = gmask[k];
    const float s2d_sum = (s2d_acc + (float)Dn * maxd) * vw;  // sum_i (min + max_d), weighted
    float d2s_sum = 0.0f, cmvw = 0.0f;
#pragma unroll
    for (int j = 0; j < Dn; j++) {
        d2s_sum += d2s[j] * cm[j];
        cmvw    += cm[j];
    }
    d2s_sum *= vw;
    cmvw    *= vw;

    // smooth L1 (beta=1) on pred_num/gt_num, masked by gt_mask
    const float diff = pnum[k] - gnum[k];
    const float ad   = fabsf(diff);
    float sl1 = (ad < 1.0f) ? 0.5f * diff * diff : (ad - 0.5f);
    sl1 *= vw;

    const float vals[5] = { s2d_sum, d2s_sum, cmvw, vw, sl1 };
#pragma unroll
    for (int s = 0; s < 5; s++) {
        const float r = block_sum(vals[s], red);
        if (threadIdx.x == 0) part[s * nblk + blockIdx.x] = r;
    }
}

// ---------- Final: deterministic reduce of partials + scalar combine ----------
__global__ void k_final(const float* __restrict__ part, int nblk, float* __restrict__ out) {
    __shared__ float red[BLOCK / 32];
    __shared__ float sums[5];
    for (int s = 0; s < 5; s++) {
        float v = 0.0f;
        for (int i = threadIdx.x; i < nblk; i += BLOCK) v += part[s * nblk + i];
        const float r = block_sum(v, red);
        if (threadIdx.x == 0) sums[s] = r;
    }
    if (threadIdx.x == 0) {
        const float s2d = sums[0], d2s = sums[1], cmvw = sums[2], vw = sums[3], sl1 = sums[4];
        const float chamfer = s2d / (vw * (float)Dn) + d2s / cmvw;
        const float numloss = sl1 / vw;
        out[0] = 1.0f * chamfer + 0.1f * numloss;   // CHAMFER_WEIGHT, PC_NUM_WEIGHT
    }
}

extern "C" void kernel_launch(void* const* d_in, const int* in_sizes, int n_in,
                              void* d_out, int out_size, void* d_ws, size_t ws_size,
                              hipStream_t stream) {
    const float* pred  = (const float*)d_in[0];  // [B,3D,H,W]
    const float* gt    = (const float*)d_in[1];  // [B,D,3,H,W]
    const float* cmask = (const float*)d_in[2];  // [B,D,H,W]
    const float* gmask = (const float*)d_in[3];  // [B,H,W]
    const float* pnum  = (const float*)d_in[4];  // [B,1,H,W]
    const float* gnum  = (const float*)d_in[5];  // [B,1,H,W]

    const int K    = in_sizes[3];        // B*H*W = 65536
    const int nblk = K / BLOCK;          // 256

    float* ws       = (float*)d_ws;
    float* blockmax = ws;                // [nblk]
    float* maxd     = ws + nblk;         // [1]
    float* part     = ws + nblk + 64;    // [5*nblk]

    k_dist_max <<<nblk, BLOCK, 0, stream>>>(pred, gt, gmask, blockmax);
    k_max_reduce<<<1,    BLOCK, 0, stream>>>(blockmax, nblk, maxd);
    k_main     <<<nblk, BLOCK, 0, stream>>>(pred, gt, cmask, gmask, pnum, gnum, maxd, nblk, part);
    k_final    <<<1,    BLOCK, 0, stream>>>(part, nblk, (float*)d_out);
}